// SelentionLinear_62878321214018
// MI455X (gfx1250) — compile-verified
//
#include <hip/hip_runtime.h>
#include <hip/hip_bf16.h>

typedef float v2f __attribute__((ext_vector_type(2)));
typedef float v8f __attribute__((ext_vector_type(8)));

#define KNOWN_POINTS 8

__device__ __forceinline__ unsigned monotonic_key(float f) {
    unsigned u = __float_as_uint(f);
    // Map float bits to an unsigned key with the same total order.
    return (u & 0x80000000u) ? ~u : (u | 0x80000000u);
}

// ---------------------------------------------------------------------------
// Kernel 1: iterative top-8 argmax over point_weight (N ~ 56k floats).
// Single workgroup of 1024 threads (32 waves). 8 passes; pass 0 pulls the
// array through L2/WGP$ (226 KB << 192 MB L2), later passes hit cache.
// Key = (monotonic(value) << 32) | ~index  ->  max == argmax, ties -> lowest
// index, matching jnp.argmax semantics in the reference.
// Outputs: sel_idx[8] and a zero-padded 16x8 A-matrix block (rows >= B are 0)
// so the GEMM kernel needs no predication.
// ---------------------------------------------------------------------------
__global__ void topk_select_kernel(const float* __restrict__ point_weight,
                                   const float* __restrict__ x,
                                   int N, int B,
                                   int*   __restrict__ sel_idx_out,  // [8]
                                   float* __restrict__ xsel_out)     // [16*8]
{
    __shared__ unsigned long long sdata[1024];
    __shared__ int selIdx[KNOWN_POINTS];
    const int tid = threadIdx.x;

    for (int k = 0; k < KNOWN_POINTS; ++k) {
        unsigned long long best = 0ull;
        for (int i = tid; i < N; i += 1024) {
            if (k == 0) {
                // gfx1250 prefetch path; pull next stride into cache.
                __builtin_prefetch(point_weight + i + 1024, 0, 3);
            }
            bool skip = false;
            #pragma unroll
            for (int j = 0; j < KNOWN_POINTS; ++j)
                skip |= (j < k) && (i == selIdx[j]);
            if (!skip) {
                unsigned long long key =
                    ((unsigned long long)monotonic_key(point_weight[i]) << 32)
                    | (unsigned)(~(unsigned)i);
                best = (key > best) ? key : best;
            }
        }
        sdata[tid] = best;
        __syncthreads();
        #pragma unroll
        for (int s = 512; s > 0; s >>= 1) {
            if (tid < s) {
                unsigned long long o = sdata[tid + s];
                if (o > sdata[tid]) sdata[tid] = o;
            }
            __syncthreads();
        }
        if (tid == 0)
            selIdx[k] = (int)~(unsigned)(sdata[0] & 0xFFFFFFFFull);
        __syncthreads();
    }

    if (tid < KNOWN_POINTS) sel_idx_out[tid] = selIdx[tid];
    if (tid < 16 * KNOWN_POINTS) {          // padded to M=16 rows
        int b = tid / KNOWN_POINTS;
        int k = tid % KNOWN_POINTS;
        xsel_out[tid] = (b < B) ? x[(long long)b * N + selIdx[k]] : 0.0f;
    }
}

// ---------------------------------------------------------------------------
// Kernel 2: out(8x256) = A(8x8) @ Wsel(8x256) + bias, exact fp32 via
// V_WMMA_F32_16X16X4_F32. One wave32 per 16-column tile of H (16 blocks).
// Per ISA f32 layouts:
//   A (16x4): lane%16 = M; VGPR0 holds K = 2*half, VGPR1 holds K = 2*half+1.
//   B (4x16): lane%16 = N; same K-pair-per-lane-half striping.
//   C/D (16x16): VGPR j = row M=j (lanes 0-15) / M=j+8 (lanes 16-31).
// Two WMMAs accumulate K=0..3 then K=4..7. No predication before the WMMAs:
// EXEC is all-ones, all loads unconditional (A is pre-padded with zeros).
// ---------------------------------------------------------------------------
__global__ void wmma_topk_gemm_kernel(const float* __restrict__ weight, // [N,H]
                                      const float* __restrict__ bias,   // [H]
                                      const int*   __restrict__ sel_idx,// [8]
                                      const float* __restrict__ xsel,   // [16*8]
                                      int H, int B,
                                      float* __restrict__ out)          // [B,H]
{
    const int lane  = threadIdx.x;   // 0..31 (wave32)
    const int n     = lane & 15;     // A row M / B col N / C col N
    const int half  = lane >> 4;     // selects K pair per ISA layout
    const int hbase = blockIdx.x * 16;
    const int col   = hbase + n;

    // Per-lane direct loads of exactly the needed K slots (no register-array
    // indexing -> no cndmask chains).
    const int kA = 2 * half;         // first WMMA:  K = kA, kA+1
    const int kB = 4 + 2 * half;     // second WMMA: K = kB, kB+1

    const int i0 = sel_idx[kA];
    const int i1 = sel_idx[kA + 1];
    const int i2 = sel_idx[kB];
    const int i3 = sel_idx[kB + 1];

    // A slots: consecutive in xsel -> one 8-byte load each (aligned).
    const v2f a0 = *(const v2f*)(xsel + n * KNOWN_POINTS + kA);
    const v2f a1 = *(const v2f*)(xsel + n * KNOWN_POINTS + kB);

    v2f b0, b1;
    b0[0] = weight[(long long)i0 * H + col];
    b0[1] = weight[(long long)i1 * H + col];
    b1[0] = weight[(long long)i2 * H + col];
    b1[1] = weight[(long long)i3 * H + col];

    const float bv = bias[col];
    v8f c;
    #pragma unroll
    for (int j = 0; j < 8; ++j) c[j] = bv;

    // (neg_a, A, neg_b, B, c_mod, C, reuse_a, reuse_b)
    c = __builtin_amdgcn_wmma_f32_16x16x4_f32(false, a0, false, b0,
                                              (short)0, c, false, false);
    c = __builtin_amdgcn_wmma_f32_16x16x4_f32(false, a1, false, b1,
                                              (short)0, c, false, false);

    if (half == 0) {
        #pragma unroll
        for (int j = 0; j < 8; ++j) {
            if (j < B) out[j * H + col] = c[j];
        }
    }
}

// ---------------------------------------------------------------------------
extern "C" void kernel_launch(void* const* d_in, const int* in_sizes, int n_in,
                              void* d_out, int out_size, void* d_ws, size_t ws_size,
                              hipStream_t stream) {
    const float* x    = (const float*)d_in[0]; // (B, N)
    const float* pw   = (const float*)d_in[1]; // (1, N)
    const float* w    = (const float*)d_in[2]; // (1, N, H)
    const float* bias = (const float*)d_in[3]; // (1, H)

    const int N = in_sizes[1];
    const int B = in_sizes[0] / N;   // 8
    const int H = in_sizes[3];       // 256

    float* out = (float*)d_out;      // (B, H)

    int*   sel_idx = (int*)d_ws;
    float* xsel    = (float*)((char*)d_ws + 64);   // 16*8 floats, 8B-aligned

    topk_select_kernel<<<1, 1024, 0, stream>>>(pw, x, N, B, sel_idx, xsel);
    wmma_topk_gemm_kernel<<<H / 16, 32, 0, stream>>>(w, bias, sel_idx, xsel,
                                                     H, B, out);
}